// GCN_88175678587120
// MI455X (gfx1250) — compile-verified
//
#include <hip/hip_runtime.h>

typedef __attribute__((ext_vector_type(2))) float v2f;
typedef __attribute__((ext_vector_type(8))) float v8f;

#define IN_SZ 256
#define HID   16
#define OUT_SZ 64

// ---------------------------------------------------------------- utilities
__global__ void fill_zero_kernel(float* __restrict__ p, long long n) {
    long long i = (long long)blockIdx.x * blockDim.x + threadIdx.x;
    if (i < n) p[i] = 0.0f;
}

__global__ void relu_inplace_kernel(float* __restrict__ p, long long n) {
    long long i = (long long)blockIdx.x * blockDim.x + threadIdx.x;
    if (i < n) p[i] = fmaxf(p[i], 0.0f);
}

// ---------------------------------------------------------------- GEMM 1
// H1[m, n] = sum_k X[m,k] * W1[n,k] + b1[n]    (m: nodes, n: 0..15, k: 0..255)
// One wave32 computes a 16x16 output tile via V_WMMA_F32_16X16X4_F32, K stepped by 4.
__global__ __launch_bounds__(256) void gemm1_wmma(const float* __restrict__ X,
                                                  const float* __restrict__ W1,
                                                  const float* __restrict__ b1,
                                                  float* __restrict__ H1,
                                                  int ntiles) {
    const int lane = threadIdx.x & 31;
    const int wave = blockIdx.x * (blockDim.x >> 5) + (threadIdx.x >> 5);
    if (wave >= ntiles) return;                 // wave-uniform: EXEC stays all-ones

    const int m0   = wave << 4;
    const int half = lane >> 4;                 // 0: K+{0,1}, 1: K+{2,3}
    const int l16  = lane & 15;

    // A fragment source: row m0+l16 of X, 2 consecutive K per lane
    const float* __restrict__ xrow = X  + (size_t)(m0 + l16) * IN_SZ + 2 * half;
    // B fragment source: B[k][n] = W1[n][k] -> row n=l16 of W1
    const float* __restrict__ wrow = W1 + (size_t)l16        * IN_SZ + 2 * half;

    v8f c = {};
    #pragma unroll 8
    for (int k0 = 0; k0 < IN_SZ; k0 += 4) {
        v2f a = { xrow[k0], xrow[k0 + 1] };
        v2f b = { wrow[k0], wrow[k0 + 1] };
        c = __builtin_amdgcn_wmma_f32_16x16x4_f32(false, a, false, b,
                                                  (short)0, c, false, false);
    }

    const float bias  = b1[l16];
    const int   mBase = m0 + 8 * half;          // C/D layout: M = r + 8*(lane>=16)
    #pragma unroll
    for (int r = 0; r < 8; ++r)
        H1[(size_t)(mBase + r) * HID + l16] = c[r] + bias;
}

// ---------------------------------------------------------------- SpMM 1 (d=16)
// S1[row[e], j] += value[e] * H1[col[e], j]   — lane-per-(edge, feature)
__global__ __launch_bounds__(256) void spmm_d16_kernel(const int* __restrict__ row,
                                                       const int* __restrict__ col,
                                                       const float* __restrict__ val,
                                                       const float* __restrict__ H1,
                                                       float* __restrict__ S1,
                                                       long long E) {
    long long t = (long long)blockIdx.x * blockDim.x + threadIdx.x;
    long long e = t >> 4;
    if (e >= E) return;
    const int j = (int)(t & 15);
    const int r = row[e];
    const int cn = col[e];
    const float v = val[e];
    atomicAdd(&S1[(size_t)r * HID + j], v * H1[(size_t)cn * HID + j]);
}

// ---------------------------------------------------------------- GEMM 2
// H2[m, n] = sum_k relu(S1[m,k]) * W2[n,k] + b2[n]   (k: 0..15, n: 0..63)
// Wave computes one 16x16 tile; 4 N-tiles per M-tile. Relu fused into A load.
__global__ __launch_bounds__(256) void gemm2_wmma(const float* __restrict__ S1,
                                                  const float* __restrict__ W2,
                                                  const float* __restrict__ b2,
                                                  float* __restrict__ H2,
                                                  int ntiles) {
    const int lane = threadIdx.x & 31;
    const int wave = blockIdx.x * (blockDim.x >> 5) + (threadIdx.x >> 5);
    if (wave >= ntiles) return;                 // wave-uniform guard

    const int mtile = wave >> 2;
    const int n0    = (wave & 3) << 4;
    const int half  = lane >> 4;
    const int l16   = lane & 15;

    const float* __restrict__ arow = S1 + (size_t)(mtile * 16 + l16) * HID + 2 * half;
    const float* __restrict__ wrow = W2 + (size_t)(n0 + l16)         * HID + 2 * half;

    v8f c = {};
    #pragma unroll
    for (int k0 = 0; k0 < HID; k0 += 4) {
        v2f a = { fmaxf(arow[k0], 0.0f), fmaxf(arow[k0 + 1], 0.0f) };
        v2f b = { wrow[k0], wrow[k0 + 1] };
        c = __builtin_amdgcn_wmma_f32_16x16x4_f32(false, a, false, b,
                                                  (short)0, c, false, false);
    }

    const float bias  = b2[n0 + l16];
    const int   mBase = mtile * 16 + 8 * half;
    #pragma unroll
    for (int r = 0; r < 8; ++r)
        H2[(size_t)(mBase + r) * OUT_SZ + n0 + l16] = c[r] + bias;
}

// ---------------------------------------------------------------- SpMM 2 (d=64)
// out[row[e], j] += value[e] * H2[col[e], j]  — lane-per-(edge, 4 features),
// float4 gathers so a 16-lane group covers one full 64-float row contiguously.
__global__ __launch_bounds__(256) void spmm_d64_kernel(const int* __restrict__ row,
                                                       const int* __restrict__ col,
                                                       const float* __restrict__ val,
                                                       const float* __restrict__ H2,
                                                       float* __restrict__ out,
                                                       long long E) {
    long long t = (long long)blockIdx.x * blockDim.x + threadIdx.x;
    long long e = t >> 4;
    if (e >= E) return;
    const int j4 = (int)(t & 15) << 2;
    const int r  = row[e];
    const int cn = col[e];
    const float v = val[e];
    const float4 h = *(const float4*)(H2 + (size_t)cn * OUT_SZ + j4);
    float* o = out + (size_t)r * OUT_SZ + j4;
    atomicAdd(o + 0, v * h.x);
    atomicAdd(o + 1, v * h.y);
    atomicAdd(o + 2, v * h.z);
    atomicAdd(o + 3, v * h.w);
}

// ---------------------------------------------------------------- launch
extern "C" void kernel_launch(void* const* d_in, const int* in_sizes, int n_in,
                              void* d_out, int out_size, void* d_ws, size_t ws_size,
                              hipStream_t stream) {
    // setup_inputs order: index, value, n, m, X, W1, b1, W2, b2
    const int*   index = (const int*)d_in[0];
    const float* value = (const float*)d_in[1];
    const float* X     = (const float*)d_in[4];
    const float* W1    = (const float*)d_in[5];
    const float* b1    = (const float*)d_in[6];
    const float* W2    = (const float*)d_in[7];
    const float* b2    = (const float*)d_in[8];

    const long long E = in_sizes[1];            // 3.2M edges
    const int       N = in_sizes[4] / IN_SZ;    // 100000 nodes

    const int* row = index;                     // index[0, :]
    const int* col = index + E;                 // index[1, :]

    // Workspace layout (38.4 MB total): H1 | S1 | H2
    float* H1 = (float*)d_ws;                   // [N, 16]
    float* S1 = H1 + (size_t)N * HID;           // [N, 16] accumulator
    float* H2 = S1 + (size_t)N * HID;           // [N, 64]
    float* out = (float*)d_out;                 // [N, 64]

    // Zero the scatter-add accumulators (d_out/d_ws are poisoned, never re-zeroed)
    {
        long long n1 = (long long)N * HID;
        fill_zero_kernel<<<(int)((n1 + 255) / 256), 256, 0, stream>>>(S1, n1);
        long long n2 = (long long)N * OUT_SZ;
        fill_zero_kernel<<<(int)((n2 + 255) / 256), 256, 0, stream>>>(out, n2);
    }

    // Layer 1 dense: H1 = X @ W1^T + b1   (wave32 per 16-row tile, WMMA f32)
    {
        int ntiles = N >> 4;                    // 6250 (N divisible by 16)
        gemm1_wmma<<<(ntiles + 7) / 8, 256, 0, stream>>>(X, W1, b1, H1, ntiles);
    }

    // Layer 1 aggregate: S1 += value * H1[col] scattered to row
    {
        long long t = E * 16;
        spmm_d16_kernel<<<(int)((t + 255) / 256), 256, 0, stream>>>(row, col, value, H1, S1, E);
    }

    // Layer 2 dense: H2 = relu(S1) @ W2^T + b2  (relu fused into A fragment)
    {
        int ntiles = (N >> 4) * 4;              // 25000 wave-tiles
        gemm2_wmma<<<(ntiles + 7) / 8, 256, 0, stream>>>(S1, W2, b2, H2, ntiles);
    }

    // Layer 2 aggregate: out += value * H2[col] scattered to row
    {
        long long t = E * 16;
        spmm_d64_kernel<<<(int)((t + 255) / 256), 256, 0, stream>>>(row, col, value, H2, out, E);
    }

    // Final relu
    {
        long long n = (long long)N * OUT_SZ;
        relu_inplace_kernel<<<(int)((n + 255) / 256), 256, 0, stream>>>(out, n);
    }
}